// WeightedOhemCELoss_78477642432764
// MI455X (gfx1250) — compile-verified
//
#include <hip/hip_runtime.h>
#include <stdint.h>

// ---------------- problem constants (fixed by the reference harness) --------
constexpr int   CLS   = 19;
constexpr int   BATCH = 8;
constexpr int   HDIM  = 512;
constexpr int   WDIM  = 1024;
constexpr int   HW    = HDIM * WDIM;          // 524288
constexpr int   NPIX  = BATCH * HW;           // 4194304
constexpr int   NBLK  = NPIX / 256;           // 16384
constexpr unsigned NMIN = 262144u;
constexpr float THRESH = 0.35667494393873245f; // -log(0.7)

// ---------------- tiny helpers ---------------------------------------------
__global__ void k_zero(unsigned* __restrict__ p, int n) {
    int i = blockIdx.x * blockDim.x + threadIdx.x;
    if (i < n) p[i] = 0u;
}

// label histogram -> counts[19]
__global__ __launch_bounds__(256) void k_label_hist(const int* __restrict__ labels,
                                                    unsigned* __restrict__ counts, int n) {
    __shared__ unsigned h[CLS];
    int tid = threadIdx.x;
    if (tid < CLS) h[tid] = 0u;
    __syncthreads();
    for (int i = blockIdx.x * blockDim.x + tid; i < n; i += gridDim.x * blockDim.x)
        atomicAdd(&h[labels[i]], 1u);
    __syncthreads();
    if (tid < CLS) atomicAdd(&counts[tid], h[tid]);
}

// ENet weights: w_c = 1 / log(1.02 + count/N)
__global__ void k_weights(const unsigned* __restrict__ counts, float* __restrict__ weights) {
    int c = threadIdx.x;
    if (c < CLS) {
        float prop = (float)counts[c] * (1.0f / (float)NPIX);
        weights[c] = 1.0f / logf(1.02f + prop);
    }
}

// ---------------- pass A: per-pixel loss via async LDS staging --------------
// Each block handles 256 consecutive pixels of one image plane. The 19x256
// fp32 logit tile (19456 B) is staged global->LDS with CDNA5 async B128
// transfers (ASYNCcnt). Logits are a 319 MB one-shot stream -> TH_LOAD_NT so
// they don't evict the reused working set (loss array, labels, hists) from L2.
__global__ __launch_bounds__(256) void k_loss(const float* __restrict__ logits,
                                              const int* __restrict__ labels,
                                              const float* __restrict__ weights,
                                              float* __restrict__ loss,
                                              unsigned* __restrict__ hist_hi) {
    __shared__ __align__(16) float tile[CLS * 256];
    __shared__ float wsh[32];

    const int tid = threadIdx.x;
    const int gi0 = blockIdx.x * 256;          // 256 | HW, so block stays in one image
    const int b   = gi0 / HW;
    const int hw0 = gi0 - b * HW;
    const float* base = logits + (size_t)b * (size_t)CLS * (size_t)HW;
    const unsigned ldsbase = (unsigned)(uintptr_t)(&tile[0]); // flat low 32 bits == LDS offset

    // 19 rows * 256 floats = 19*64 chunks of 16B
    #pragma unroll
    for (int it = 0; it < 5; ++it) {
        int t = it * 256 + tid;
        if (t < CLS * 64) {
            int c = t >> 6;
            int q = t & 63;
            unsigned voff = (unsigned)((c * HW + hw0) * 4 + q * 16); // byte offset vs base
            unsigned ldst = ldsbase + (unsigned)t * 16u;
            asm volatile("global_load_async_to_lds_b128 %0, %1, %2 th:TH_LOAD_NT"
                         :: "v"(ldst), "v"(voff), "s"(base) : "memory");
        }
    }
    if (tid < CLS) wsh[tid] = weights[tid];

    asm volatile("s_wait_asynccnt 0" ::: "memory");  // my wave's async transfers done
    __syncthreads();                                  // all waves' transfers visible

    float x[CLS];
    #pragma unroll
    for (int c = 0; c < CLS; ++c) x[c] = tile[c * 256 + tid];
    float m = x[0];
    #pragma unroll
    for (int c = 1; c < CLS; ++c) m = fmaxf(m, x[c]);
    float s = 0.0f;
    #pragma unroll
    for (int c = 0; c < CLS; ++c) s += __expf(x[c] - m);

    const int lab = labels[gi0 + tid];
    const float xl = tile[lab * 256 + tid];
    const float l  = wsh[lab] * ((m - xl) + __logf(s));  // nonnegative by construction

    loss[gi0 + tid] = l;
    atomicAdd(&hist_hi[__float_as_uint(l) >> 16], 1u);
}

// ---------------- parallel pivot search over hi-bit histogram ----------------
// 1024 threads, 64 buckets each: chunk sums -> LDS suffix scan (10 steps) ->
// exactly one thread owns the crossing chunk and walks its 64 buckets.
// scal[0]=B, scal[1]=count above, scal[2]=r, scal[3]=cntGt (filled by pass B)
__global__ __launch_bounds__(1024) void k_scan_hi(const unsigned* __restrict__ hist_hi,
                                                  unsigned* __restrict__ scal) {
    __shared__ unsigned s[1024];
    const int tid  = threadIdx.x;
    const int base = tid * 64;
    unsigned local = 0;
    #pragma unroll 8
    for (int k = 0; k < 64; ++k) local += hist_hi[base + k];
    s[tid] = local;
    __syncthreads();
    // inclusive suffix scan over chunk sums (Hillis-Steele)
    for (int ofs = 1; ofs < 1024; ofs <<= 1) {
        unsigned v   = s[tid];
        unsigned add = (tid + ofs < 1024) ? s[tid + ofs] : 0u;
        __syncthreads();
        s[tid] = v + add;
        __syncthreads();
    }
    const unsigned A = (tid < 1023) ? s[tid + 1] : 0u;   // count in buckets above my chunk
    if (A < NMIN && A + local >= NMIN) {                  // unique crossing chunk
        unsigned cum = A;
        for (int j = base + 63; j >= base; --j) {
            unsigned h = hist_hi[j];
            if (cum + h >= NMIN) {
                scal[0] = (unsigned)j; scal[1] = cum; scal[2] = NMIN - cum;
                break;
            }
            cum += h;
        }
    }
}

// ---------------- pass B: exact sums + low-bit histogram inside pivot -------
__global__ __launch_bounds__(256) void k_pass2(const float* __restrict__ loss,
                                               unsigned* __restrict__ scal,
                                               unsigned* __restrict__ hist_lo,
                                               float* __restrict__ blkSumAbove,
                                               float* __restrict__ blkSumGt) {
    __shared__ float r1[256];
    __shared__ float r2[256];
    __shared__ unsigned r3[256];
    const int tid = threadIdx.x;
    const int gi  = blockIdx.x * 256 + tid;
    const unsigned B = scal[0];

    const float l = __builtin_nontemporal_load(&loss[gi]);  // last consumer of loss[]
    const unsigned bits = __float_as_uint(l);
    const unsigned hi = bits >> 16;

    float sA = (hi > B) ? l : 0.0f;
    if (hi == B) atomicAdd(&hist_lo[bits & 0xFFFFu], 1u);
    float sG = (l > THRESH) ? l : 0.0f;
    unsigned cG = (l > THRESH) ? 1u : 0u;

    r1[tid] = sA; r2[tid] = sG; r3[tid] = cG;
    __syncthreads();
    for (int ofs = 128; ofs > 0; ofs >>= 1) {
        if (tid < ofs) { r1[tid] += r1[tid + ofs]; r2[tid] += r2[tid + ofs]; r3[tid] += r3[tid + ofs]; }
        __syncthreads();
    }
    if (tid == 0) {
        blkSumAbove[blockIdx.x] = r1[0];
        blkSumGt[blockIdx.x]    = r2[0];
        atomicAdd(&scal[3], r3[0]);   // integer atomic: deterministic
    }
}

// ---------------- final: deterministic reduce + parallel exact top-r --------
// 1024 threads: (a) fixed-order reduce of per-block partials, (b) chunk-suffix
// scan of hist_lo, per-bucket take = clamp(r - cumBefore, 0, cnt) -> exact
// top-r sum inside pivot bucket (every (hi,lo) pair is one exact float value).
__global__ __launch_bounds__(1024) void k_final(const float* __restrict__ blkSumAbove,
                                                const float* __restrict__ blkSumGt,
                                                const unsigned* __restrict__ scal,
                                                const unsigned* __restrict__ hist_lo,
                                                float* __restrict__ out, int nblk) {
    __shared__ unsigned sc[1024];
    __shared__ float r1[1024];
    __shared__ float r2[1024];
    __shared__ float r3[1024];
    const int tid = threadIdx.x;

    // (a) per-block partial sums, fixed order
    float a = 0.0f, g = 0.0f;
    for (int i = tid; i < nblk; i += 1024) { a += blkSumAbove[i]; g += blkSumGt[i]; }

    // (b) chunk sums of hist_lo + suffix scan
    const unsigned B = scal[0];
    const unsigned r = scal[2];
    const int base = tid * 64;
    unsigned local = 0;
    #pragma unroll 8
    for (int k = 0; k < 64; ++k) local += hist_lo[base + k];
    sc[tid] = local;
    __syncthreads();
    for (int ofs = 1; ofs < 1024; ofs <<= 1) {
        unsigned v   = sc[tid];
        unsigned add = (tid + ofs < 1024) ? sc[tid + ofs] : 0u;
        __syncthreads();
        sc[tid] = v + add;
        __syncthreads();
    }
    const unsigned A = (tid < 1023) ? sc[tid + 1] : 0u;  // elements in lo-buckets above my chunk

    float sel = 0.0f;
    if (A < r && local > 0) {
        unsigned cum = A;
        for (int j = base + 63; j >= base && cum < r; --j) {
            unsigned c = hist_lo[j];
            if (!c) continue;
            unsigned rem  = r - cum;
            unsigned take = (c < rem) ? c : rem;
            sel += (float)take * __uint_as_float((B << 16) | (unsigned)j);
            cum += take;
        }
    }

    r1[tid] = a; r2[tid] = g; r3[tid] = sel;
    __syncthreads();
    for (int ofs = 512; ofs > 0; ofs >>= 1) {
        if (tid < ofs) { r1[tid] += r1[tid + ofs]; r2[tid] += r2[tid + ofs]; r3[tid] += r3[tid + ofs]; }
        __syncthreads();
    }
    if (tid == 0) {
        const unsigned cg = scal[3];
        float res = (cg > NMIN) ? (r2[0] / (float)cg)
                                : ((r1[0] + r3[0]) / (float)NMIN);
        out[0] = res;
    }
}

// ---------------- launcher ---------------------------------------------------
extern "C" void kernel_launch(void* const* d_in, const int* in_sizes, int n_in,
                              void* d_out, int out_size, void* d_ws, size_t ws_size,
                              hipStream_t stream) {
    (void)in_sizes; (void)n_in; (void)out_size; (void)ws_size;
    const float* logits = (const float*)d_in[0];
    const int*   labels = (const int*)d_in[1];
    float*       out    = (float*)d_out;

    // workspace carve-up (~16.7 MB total)
    float*    loss        = (float*)d_ws;                 // NPIX f32
    unsigned* hist_hi     = (unsigned*)(loss + NPIX);     // 65536 u32
    unsigned* hist_lo     = hist_hi + 65536;              // 65536 u32
    unsigned* counts      = hist_lo + 65536;              // 32 u32
    float*    weights     = (float*)(counts + 32);        // 32 f32
    unsigned* scal        = (unsigned*)(weights + 32);    // 16 u32
    float*    blkSumGt    = (float*)(scal + 16);          // NBLK f32
    float*    blkSumAbove = blkSumGt + NBLK;              // NBLK f32

    // zero everything we accumulate into (contiguous: hist_hi..scal)
    const int zn = 65536 + 65536 + 32 + 32 + 16;
    k_zero<<<(zn + 255) / 256, 256, 0, stream>>>(hist_hi, zn);

    k_label_hist<<<1024, 256, 0, stream>>>(labels, counts, NPIX);
    k_weights<<<1, 32, 0, stream>>>(counts, weights);
    k_loss<<<NBLK, 256, 0, stream>>>(logits, labels, weights, loss, hist_hi);
    k_scan_hi<<<1, 1024, 0, stream>>>(hist_hi, scal);
    k_pass2<<<NBLK, 256, 0, stream>>>(loss, scal, hist_lo, blkSumAbove, blkSumGt);
    k_final<<<1, 1024, 0, stream>>>(blkSumAbove, blkSumGt, scal, hist_lo, out, NBLK);
}